// GDMLTorchAssemble_13443247636896
// MI455X (gfx1250) — compile-verified
//
#include <hip/hip_runtime.h>
#include <math.h>

#define N_TRAIN 2000
#define N_ATOMS 21
#define DIM_D   210          // pairwise descriptors
#define DIM_I   63           // 3*N_ATOMS
#define N_PERMS 12
#define N_COLS  8
#define DP      212          // DIM_D padded to multiple of 4 (WMMA K-step)
#define DPAD    224          // DIM_D padded to multiple of 16 (outer1 M tiles)
#define DSTRIDE 226          // row stride for d-contiguous layouts (bank-conflict-free)
#define IP      64           // DIM_I padded to 64 (4 x 16 tiles)
#define SIG     10.0f

typedef float v2f __attribute__((ext_vector_type(2)));
typedef float v8f __attribute__((ext_vector_type(8)));

// Dynamic LDS (~154 KB -> two workgroups co-resident per 320 KB WGP)
struct Smem {
    float rdn[DIM_D];              // R_desc[n]
    float rdj[DIM_D];              // R_desc[jn]
    float dn[DIM_D * 3];           // compact Jacobian of point n
    float dj[DIM_D * 3];           // compact Jacobian of point jn
    short ka[DIM_D];               // tril row atom (a > b)
    short kb[DIM_D];               // tril col atom
    unsigned short pidx [N_PERMS * DIM_D];  // perm_idx[p][d]
    unsigned short ipidx[N_PERMS * DIM_D];  // inverse perm: d s.t. pidx[p][d]=k
    float diffp[N_PERMS * DP];     // diff[p][k] (permuted to j's k-order)
    float bmat[N_PERMS * IP];      // b[p][e]
    float qv[N_PERMS];             // 5 * mat52[p]
    float cv[N_PERMS];             // (SIG^2 + SIG*norm) * mat52[p]
    alignas(8) float qd[DPAD * N_PERMS];     // qv[p]*diff[p][d], [d][p], zero padded
    alignas(8) float RN_T [IP * DSTRIDE];    // Rd_full[n]^T : [m][d], d-contiguous
    alignas(8) float OUT_T[IP * DSTRIDE];    // outer^T      : [e][d], d-contiguous
};

__global__ __launch_bounds__(256)
void gdml_assemble_kernel(const float* __restrict__ R_desc,
                          const float* __restrict__ R_d_desc,
                          const int*   __restrict__ tril_perms_lin,
                          const int*   __restrict__ j_idxs,
                          float* __restrict__ out)
{
    extern __shared__ char raw_smem[];
    Smem& S = *reinterpret_cast<Smem*>(raw_smem);

    const int jcol = blockIdx.x;          // 0..7
    const int n    = blockIdx.y;          // 0..1999
    const int t    = threadIdx.x;         // 0..255
    const int jn   = j_idxs[jcol];

    const int wave = t >> 5;
    const int lane = t & 31;
    const int half = lane >> 4;           // lane half selects K pair
    const int lmod = lane & 15;           // m (A) / n (B) within a 16x16 tile

    // ---- Phase 1: stage inputs -------------------------------------------
    for (int i = t; i < DIM_D; i += 256) {
        S.rdn[i] = R_desc[(size_t)n  * DIM_D + i];
        S.rdj[i] = R_desc[(size_t)jn * DIM_D + i];
        int a = 1;                         // decode tril pair (a,b), a > b
        while ((a * (a + 1)) / 2 <= i) a++;
        S.ka[i] = (short)a;
        S.kb[i] = (short)(i - (a * (a - 1)) / 2);
    }
    for (int i = t; i < DIM_D * 3; i += 256) {
        S.dn[i] = R_d_desc[(size_t)n  * DIM_D * 3 + i];
        S.dj[i] = R_d_desc[(size_t)jn * DIM_D * 3 + i];
    }
    for (int i = t; i < N_PERMS * DIM_D; i += 256) {
        int p = i / DIM_D, d = i % DIM_D;
        S.pidx[p * DIM_D + d] =
            (unsigned short)(tril_perms_lin[d * N_PERMS + p] % DIM_D);
    }
    __syncthreads();

    // ---- Phase 2: diffp (k-order), inverse perm, dense RN^T --------------
    for (int i = t; i < N_PERMS * DIM_D; i += 256) {
        int p = i / DIM_D, d = i % DIM_D;
        int kd = S.pidx[p * DIM_D + d];
        S.diffp[p * DP + kd]    = S.rdn[d] - S.rdj[kd];
        S.ipidx[p * DIM_D + kd] = (unsigned short)d;
    }
    // RN^T[m][d] = Rd_full[n][d][m]  (6 nonzeros per d-row)
    for (int i = t; i < IP * DSTRIDE; i += 256) {
        int m = i / DSTRIDE, d = i % DSTRIDE;
        float v = 0.0f;
        if (d < DIM_D && m < DIM_I) {
            int at = m / 3, comp = m % 3;
            if      (at == S.ka[d]) v =  S.dn[d * 3 + comp];
            else if (at == S.kb[d]) v = -S.dn[d * 3 + comp];
        }
        S.RN_T[i] = v;
    }
    __syncthreads();

    // ---- Phase 3: per-permutation scalars (norm is perm-invariant sum) ---
    if (t < N_PERMS) {
        float s = 0.0f;
        for (int k = 0; k < DIM_D; ++k) {
            float v = S.diffp[t * DP + k];
            s += v * v;
        }
        const float sqrt5 = 2.2360679774997896f;
        float norm  = sqrt5 * sqrtf(s);
        float mat52 = expf(-norm / SIG) * (5.0f / (3.0f * SIG * SIG * SIG * SIG));
        S.qv[t] = 5.0f * mat52;
        S.cv[t] = (SIG * SIG + SIG * norm) * mat52;
    }
    __syncthreads();

    // ---- Phase 4: b[p][e] (sparse, 20 pairs per atom) and qd[d][p] -------
    for (int i = t; i < N_PERMS * DIM_I; i += 256) {
        int p = i / DIM_I, e = i % DIM_I;
        int at = e / 3, comp = e % 3;
        float acc = 0.0f;
        int kbase = (at * (at - 1)) / 2;
        for (int s = 0; s < at; ++s) {                // `at` is tril row: +
            int k = kbase + s;
            acc += S.diffp[p * DP + k] * S.dj[k * 3 + comp];
        }
        for (int s = at + 1; s < N_ATOMS; ++s) {      // `at` is tril col: -
            int k = (s * (s - 1)) / 2 + at;
            acc -= S.diffp[p * DP + k] * S.dj[k * 3 + comp];
        }
        S.bmat[p * IP + e] = acc;
    }
    // qd[d][p] = qv[p] * diff[p][d] ; diff[p][d] == diffp[p][pidx[p][d]]
    for (int i = t; i < DPAD * N_PERMS; i += 256) {
        int d = i / N_PERMS, p = i % N_PERMS;
        float v = 0.0f;
        if (d < DIM_D) {
            int kd = S.pidx[p * DIM_D + d];
            v = S.qv[p] * S.diffp[p * DP + kd];
        }
        S.qd[i] = v;
    }
    __syncthreads();

    // ---- Phase 5: outer1 = qd(224x12) x b(12x64) via WMMA -> OUT_T -------
    // 14 x 4 = 56 tiles of 16x16, K = 12 = 3 chunks of 4; 8 waves, 7 tiles each
    for (int tile = wave; tile < 14 * 4; tile += 8) {
        int td = tile >> 2;               // d tile
        int te = tile & 3;                // e tile
        v8f acc = {};
        #pragma unroll
        for (int kk = 0; kk < 3; ++kk) {
            int k0 = kk * 4 + 2 * half;
            v2f a = *reinterpret_cast<const v2f*>(
                        &S.qd[(td * 16 + lmod) * N_PERMS + k0]);
            v2f b;
            b.x = S.bmat[(k0 + 0) * IP + te * 16 + lmod];
            b.y = S.bmat[(k0 + 1) * IP + te * 16 + lmod];
            acc = __builtin_amdgcn_wmma_f32_16x16x4_f32(
                      false, a, false, b, (short)0, acc, false, false);
        }
        #pragma unroll
        for (int v = 0; v < 8; ++v) {     // D: VGPR v -> M = v + 8*half, N = lmod
            int drow = td * 16 + v + 8 * half;
            int ecol = te * 16 + lmod;
            S.OUT_T[ecol * DSTRIDE + drow] = acc[v];
        }
    }
    __syncthreads();

    // ---- Phase 6: outer2 scatter: OUT_T[e][d] -= cv[p] * RdFull[j][k][e] --
    // d = ipidx[p][k]; Jacobian row k has 6 nonzeros -> 15K ds_add_f32
    for (int i = t; i < N_PERMS * DIM_D; i += 256) {
        int p = i / DIM_D, k = i % DIM_D;
        int d  = S.ipidx[p * DIM_D + k];
        int ea = 3 * (int)S.ka[k], eb = 3 * (int)S.kb[k];
        float cp = S.cv[p];
        #pragma unroll
        for (int comp = 0; comp < 3; ++comp) {
            float v = cp * S.dj[k * 3 + comp];
            atomicAdd(&S.OUT_T[(ea + comp) * DSTRIDE + d], -v);
            atomicAdd(&S.OUT_T[(eb + comp) * DSTRIDE + d],  v);
        }
    }
    __syncthreads();

    // ---- Phase 7: kblk = RN^T(63x210) x outer(210x63) via WMMA -----------
    // 64x64 = 4x4 tiles; wave w: row tile w>>1, col tiles {0,1}+2*(w&1);
    // A fragment shared by both column tiles; K = 212 = 53 chunks of 4
    {
        const int tm  = wave >> 1;
        const int te0 = (wave & 1) * 2;
        v8f acc0 = {};
        v8f acc1 = {};
        for (int kk = 0; kk < DP / 4; ++kk) {
            int k0 = kk * 4 + 2 * half;
            v2f a  = *reinterpret_cast<const v2f*>(
                         &S.RN_T[(tm * 16 + lmod) * DSTRIDE + k0]);
            v2f b0 = *reinterpret_cast<const v2f*>(
                         &S.OUT_T[((te0 + 0) * 16 + lmod) * DSTRIDE + k0]);
            v2f b1 = *reinterpret_cast<const v2f*>(
                         &S.OUT_T[((te0 + 1) * 16 + lmod) * DSTRIDE + k0]);
            acc0 = __builtin_amdgcn_wmma_f32_16x16x4_f32(
                       false, a, false, b0, (short)0, acc0, false, false);
            acc1 = __builtin_amdgcn_wmma_f32_16x16x4_f32(
                       false, a, false, b1, (short)0, acc1, false, false);
        }
        #pragma unroll
        for (int v = 0; v < 8; ++v) {
            int krow = tm * 16 + v + 8 * half;
            if (krow < DIM_I) {
                size_t row = ((size_t)n * DIM_I + krow) * (N_COLS * DIM_I)
                           + (size_t)jcol * DIM_I;
                int e0 = (te0 + 0) * 16 + lmod;
                int e1 = (te0 + 1) * 16 + lmod;
                if (e0 < DIM_I) out[row + e0] = acc0[v];
                if (e1 < DIM_I) out[row + e1] = acc1[v];
            }
        }
    }
}

extern "C" void kernel_launch(void* const* d_in, const int* in_sizes, int n_in,
                              void* d_out, int out_size, void* d_ws, size_t ws_size,
                              hipStream_t stream) {
    const float* R_desc         = (const float*)d_in[0];
    const float* R_d_desc       = (const float*)d_in[1];
    const int*   tril_perms_lin = (const int*)d_in[2];
    const int*   j_idxs         = (const int*)d_in[3];
    float* out = (float*)d_out;

    // allow >64KB dynamic LDS (no-op where not required; not a stream op)
    (void)hipFuncSetAttribute((const void*)gdml_assemble_kernel,
                              hipFuncAttributeMaxDynamicSharedMemorySize,
                              (int)sizeof(Smem));

    dim3 grid(N_COLS, N_TRAIN);
    size_t smem = sizeof(Smem);
    gdml_assemble_kernel<<<grid, 256, smem, stream>>>(
        R_desc, R_d_desc, tril_perms_lin, j_idxs, out);
}